// HGCNPlus_5007931867343
// MI455X (gfx1250) — compile-verified
//
#include <hip/hip_runtime.h>
#include <math.h>

#define N_NODES 50000
#define E_EDGES 800000
#define HID 128
#define OUT_DIM 64
#define EPSV 1e-12f
#define MAXNF 0.99999f   // 1.0 - 1e-5

typedef __attribute__((ext_vector_type(2))) float v2f;
typedef __attribute__((ext_vector_type(8))) float v8f;

// ---------------- utility kernels ----------------

__global__ void k_zero(float* __restrict__ p, int n) {
    int i = blockIdx.x * blockDim.x + threadIdx.x;
    if (i < n) p[i] = 0.0f;
}

__global__ void k_count_deg(const int* __restrict__ dst, float* __restrict__ deg, int e) {
    int i = blockIdx.x * blockDim.x + threadIdx.x;
    if (i < e) atomicAdd(&deg[dst[i]], 1.0f);
}

__global__ void k_invdeg(float* __restrict__ deg, int n) {
    int i = blockIdx.x * blockDim.x + threadIdx.x;
    if (i < n) deg[i] = 1.0f / fmaxf(deg[i], 1.0f);
}

// ---------------- row-wise hyperbolic maps ----------------
// one wave (32 lanes) per 128-dim row; 8 rows per 256-thread block

// alpha[r] = arctanh(sc*nclip)/(sc*nclip), nclip = clamp(||h_r||, EPS, MAXNF/sc)
__global__ void k_alpha(const float* __restrict__ h, float* __restrict__ alpha,
                        const float* __restrict__ cp) {
    int row  = blockIdx.x * 8 + (threadIdx.x >> 5);
    int lane = threadIdx.x & 31;
    float sc = sqrtf(fabsf(*cp));
    float4 v = ((const float4*)(h + (size_t)row * HID))[lane];
    float s = v.x * v.x + v.y * v.y + v.z * v.z + v.w * v.w;
    #pragma unroll
    for (int m = 16; m >= 1; m >>= 1) s += __shfl_xor(s, m, 32);
    if (lane == 0) {
        float n  = sqrtf(s);
        float nc = fminf(fmaxf(n, EPSV), MAXNF / sc);
        alpha[row] = atanhf(sc * nc) / (sc * nc);
    }
}

// out_r = tanh(sc*n)/(sc*n) * v,  v = (invdeg ? invdeg[r] : 1) * in_r,  n = max(||v||, EPS)
// safe in-place (each lane reads then writes its own float4)
__global__ void k_expmap(const float* __restrict__ in, float* __restrict__ out,
                         const float* __restrict__ cp, const float* __restrict__ invdeg) {
    int row  = blockIdx.x * 8 + (threadIdx.x >> 5);
    int lane = threadIdx.x & 31;
    float sc = sqrtf(fabsf(*cp));
    float m = invdeg ? invdeg[row] : 1.0f;
    float4 v = ((const float4*)(in + (size_t)row * HID))[lane];
    v.x *= m; v.y *= m; v.z *= m; v.w *= m;
    float s = v.x * v.x + v.y * v.y + v.z * v.z + v.w * v.w;
    #pragma unroll
    for (int mm = 16; mm >= 1; mm >>= 1) s += __shfl_xor(s, mm, 32);
    float n = fmaxf(sqrtf(s), EPSV);
    float scl = tanhf(sc * n) / (sc * n);
    v.x *= scl; v.y *= scl; v.z *= scl; v.w *= scl;
    ((float4*)(out + (size_t)row * HID))[lane] = v;
}

// ---------------- edge scatter (mean aggregation numerator) ----------------
// 1 thread per (edge, dim): coalesced gather of t[src], f32 atomic add into agg[dst]
__global__ void k_scatter(const float* __restrict__ t, const int* __restrict__ src,
                          const int* __restrict__ dst, float* __restrict__ agg) {
    long long tid = (long long)blockIdx.x * blockDim.x + threadIdx.x;
    int e = (int)(tid >> 7);
    int d = (int)(tid & 127);
    if (e < E_EDGES) {
        int s  = src[e];
        int dd = dst[e];
        atomicAdd(&agg[(size_t)dd * HID + d], t[(size_t)s * HID + d]);
    }
}

// ---------------- fused fp32 WMMA GEMM ----------------
// out[r][c] = (alpha ? alpha[r] : 1) * sum_k A[r][k]*W[k][c] + bias[c]
// A: [*,128] row-major, W: [128,NCOLS] row-major.
// blockDim = 32*(NCOLS/16); each wave owns one 16x16 tile; block covers 16 full rows.
template <int NCOLS>
__global__ void k_gemm(const float* __restrict__ A, const float* __restrict__ W,
                       const float* __restrict__ bias, const float* __restrict__ alpha,
                       float* __restrict__ out) {
    const int lane = threadIdx.x & 31;
    const int wv   = threadIdx.x >> 5;    // column-tile index
    const int hi   = lane >> 4;           // 0: K pair {k,k+1}, 1: {k+2,k+3}
    const int r    = lane & 15;
    const int row0 = blockIdx.x * 16;
    const int col  = (wv << 4) + r;

    const float* Ap = A + (size_t)(row0 + r) * HID + 2 * hi;   // 8B aligned
    const float* Bp = W + (size_t)(2 * hi) * NCOLS + col;

    v8f acc = {};
    #pragma unroll
    for (int k = 0; k < HID; k += 4) {
        v2f a = *(const v2f*)Ap;          // A[r][k+2hi], A[r][k+2hi+1]
        v2f b;
        b[0] = Bp[0];                     // W[k+2hi][col]
        b[1] = Bp[NCOLS];                 // W[k+2hi+1][col]
        acc = __builtin_amdgcn_wmma_f32_16x16x4_f32(
            false, a, false, b, (short)0, acc, false, false);
        Ap += 4;
        Bp += 4 * NCOLS;
    }

    float bb = bias[col];
    #pragma unroll
    for (int i = 0; i < 8; ++i) {
        int rr = row0 + i + 8 * hi;       // D layout: VGPR i -> M = i + 8*(lane>=16)
        float v = acc[i];
        if (alpha) v *= alpha[rr];
        out[(size_t)rr * NCOLS + col] = v + bb;
    }
}

// ---------------- launch ----------------

extern "C" void kernel_launch(void* const* d_in, const int* in_sizes, int n_in,
                              void* d_out, int out_size, void* d_ws, size_t ws_size,
                              hipStream_t stream) {
    (void)in_sizes; (void)n_in; (void)out_size; (void)ws_size;
    const float* x      = (const float*)d_in[0];
    const int*   ei     = (const int*)d_in[1];     // [2, E]
    const float* cp     = (const float*)d_in[2];
    const float* enc_w  = (const float*)d_in[3];
    const float* enc_b  = (const float*)d_in[4];
    const float* conv_w = (const float*)d_in[5];   // [3,128,128]
    const float* conv_b = (const float*)d_in[6];   // [3,128]
    const float* head_w = (const float*)d_in[7];   // [128,64]
    const float* head_b = (const float*)d_in[8];
    float* outp = (float*)d_out;

    float* t      = (float*)d_ws;                       // N*128
    float* h      = t + (size_t)N_NODES * HID;          // N*128
    float* invdeg = h + (size_t)N_NODES * HID;          // N (deg, then 1/max(deg,1))
    float* alpha  = invdeg + N_NODES;                   // N

    const int* src = ei;
    const int* dst = ei + E_EDGES;

    // degree -> inverse degree
    k_zero<<<(N_NODES + 255) / 256, 256, 0, stream>>>(invdeg, N_NODES);
    k_count_deg<<<(E_EDGES + 255) / 256, 256, 0, stream>>>(dst, invdeg, E_EDGES);
    k_invdeg<<<(N_NODES + 255) / 256, 256, 0, stream>>>(invdeg, N_NODES);

    // encoder: u = x @ enc_w + enc_b ; h = expmap0(u)
    k_gemm<HID><<<N_NODES / 16, 256, 0, stream>>>(x, enc_w, enc_b, nullptr, t);
    k_expmap<<<N_NODES / 8, 256, 0, stream>>>(t, h, cp, nullptr);

    for (int l = 0; l < 3; ++l) {
        // t = (alpha ⊙ h) @ conv_w[l] + conv_b[l]   (logmap folded into epilogue)
        k_alpha<<<N_NODES / 8, 256, 0, stream>>>(h, alpha, cp);
        k_gemm<HID><<<N_NODES / 16, 256, 0, stream>>>(
            h, conv_w + (size_t)l * HID * HID, conv_b + (size_t)l * HID, alpha, t);
        // h <- segment_sum(t[src], dst); then h = expmap0(inv_deg * h)
        k_zero<<<(N_NODES * HID + 255) / 256, 256, 0, stream>>>(h, N_NODES * HID);
        k_scatter<<<((long long)E_EDGES * HID) / 256, 256, 0, stream>>>(t, src, dst, h);
        k_expmap<<<N_NODES / 8, 256, 0, stream>>>(h, h, cp, invdeg);
    }

    // head: out = (alpha ⊙ h) @ head_w + head_b
    k_alpha<<<N_NODES / 8, 256, 0, stream>>>(h, alpha, cp);
    k_gemm<OUT_DIM><<<N_NODES / 16, 128, 0, stream>>>(h, head_w, head_b, alpha, outp);
}